// DynamicPatcher_62448824484363
// MI455X (gfx1250) — compile-verified
//
#include <hip/hip_runtime.h>
#include <hip/hip_bf16.h>
#include <stddef.h>

// DynamicPatcher for MI455X (gfx1250, wave32).
//
// Outputs are input-independent (entropy is never read by the reference).
//   patch_lengths[b][p] : static fill loop over (S, P=1024)
//   patch_ids[b][s]     = searchsorted_right(cumsum(lengths), s)
//                       = min(s / avg, min((S-1)/avg, P-1))   (closed form)
//
// Pure-store workload: 67.1 MB of int32 writes, zero loads required.
// Store-bandwidth bound: ~2.9 us at 23.3 TB/s. No matrix ops exist, so
// WMMA/TDM are inapplicable; optimal path = coalesced wave32 b128 NT stores
// with the patch-id math hoisted to SALU (uniform per block for this shape).

#define MAX_PATCHES 1024
#define THREADS     256                   // 8 wave32 per block
#define BLOCK_ELEMS (THREADS * 4)         // int32 elements per block (b128/thread)

typedef __attribute__((ext_vector_type(4))) int int4v;

// ---------------------------------------------------------------------------
// patch_lengths: one thread per element (tiny: B*P = 16K elements).
// Closed-form emulation of the reference's sequential fill loop, including
// the quirk that a mid-loop break leaves `remaining` unchanged and it is
// ALSO written into the last slot.
// ---------------------------------------------------------------------------
__global__ void dp_lengths_kernel(int* __restrict__ out, int total, int P,
                                  int avg, int pb, int S) {
    int i = blockIdx.x * blockDim.x + threadIdx.x;
    if (i >= total) return;
    int p = i % P;
    int pbc = pb < P - 1 ? pb : P - 1;   // index of first non-avg slot (capped)
    int len;
    if (p < pbc) {
        len = avg;                        // full patches before the break
    } else if (pb <= P - 2 && p == pb) {
        len = S - pb * avg;               // break writes the remainder here
    } else if (p == P - 1) {
        int rem = (pb <= P - 2) ? (S - pb * avg)        // quirk: remainder again
                                : (S - (P - 1) * avg);  // no break: leftover
        len = rem > 0 ? rem : 0;
    } else {
        len = 0;                          // slots after a break stay zero
    }
    out[i] = len;
}

// ---------------------------------------------------------------------------
// patch_ids, fastest path: avg % BLOCK_ELEMS == 0  (this instance: avg=1024).
// Every 256-thread block covers 1024 consecutive positions inside ONE patch,
// so the id is uniform per block -> computed once in SALU from blockIdx.x.
// Per thread: address shift + one non-temporal global_store_b128.
// ---------------------------------------------------------------------------
__global__ void dp_ids_uniform_kernel(int4v* __restrict__ out, unsigned s_groups,
                                      unsigned blocks_per_patch, unsigned pbc) {
    unsigned g = blockIdx.x * THREADS + threadIdx.x;      // b128 group in row
    if (g >= s_groups) return;
    unsigned q = blockIdx.x / blocks_per_patch;           // scalar (SALU) divide
    int v = (int)(q < pbc ? q : pbc);
    int4v val = {v, v, v, v};
    __builtin_nontemporal_store(val, out + (size_t)blockIdx.y * s_groups + g);
}

// ---------------------------------------------------------------------------
// patch_ids, pow2-avg path: v = min(s >> sh, pbc); 2 VALU + 1 store / thread.
// Requires avg power of two, avg % 4 == 0, S % 4 == 0.
// ---------------------------------------------------------------------------
__global__ void dp_ids_shift_kernel(int4v* __restrict__ out, unsigned s_groups,
                                    unsigned sh, unsigned pbc) {
    unsigned g = blockIdx.x * THREADS + threadIdx.x;
    if (g >= s_groups) return;
    unsigned q = (g * 4u) >> sh;
    int v = (int)(q < pbc ? q : pbc);
    int4v val = {v, v, v, v};
    __builtin_nontemporal_store(val, out + (size_t)blockIdx.y * s_groups + g);
}

// ---------------------------------------------------------------------------
// patch_ids, general grouped path: runtime division (magic expansion).
// Requires avg % 4 == 0 and S % 4 == 0.
// ---------------------------------------------------------------------------
__global__ void dp_ids_b128_kernel(int4v* __restrict__ out, unsigned s_groups,
                                   unsigned avg, unsigned pbc) {
    unsigned g = blockIdx.x * THREADS + threadIdx.x;
    if (g >= s_groups) return;
    unsigned q = (g * 4u) / avg;
    int v = (int)(q < pbc ? q : pbc);
    int4v val = {v, v, v, v};
    __builtin_nontemporal_store(val, out + (size_t)blockIdx.y * s_groups + g);
}

// ---------------------------------------------------------------------------
// patch_ids, fully general fallback: one int32 per thread.
// ---------------------------------------------------------------------------
__global__ void dp_ids_scalar_kernel(int* __restrict__ out, unsigned total,
                                     unsigned S, unsigned avg, unsigned pbc) {
    unsigned i = blockIdx.x * blockDim.x + threadIdx.x;
    if (i >= total) return;
    unsigned q = (i % S) / avg;
    out[i] = (int)(q < pbc ? q : pbc);
}

extern "C" void kernel_launch(void* const* d_in, const int* in_sizes, int n_in,
                              void* d_out, int out_size, void* d_ws, size_t ws_size,
                              hipStream_t stream) {
    (void)d_in; (void)n_in; (void)d_ws; (void)ws_size;

    // Derive shapes: in = B*S elements, out = B*P + B*S elements, P = 1024.
    const long long P  = MAX_PATCHES;
    const long long BS = in_sizes[0];
    long long B = ((long long)out_size - BS) / P;
    if (B <= 0) B = 1;
    const long long S = BS / B;

    // Reference's static loop parameters (closed form).
    long long avg = S / P; if (avg < 1) avg = 1;
    long long pb  = (S - 1) / avg;                 // first index with rem <= avg
    long long pbc = pb < P - 1 ? pb : P - 1;       // id cap

    int* out_len = (int*)d_out;                    // B*P int32
    int* out_ids = out_len + (size_t)(B * P);      // B*S int32 (64 KiB offset,
                                                   //  16B-aligned)

    // ---- patch_lengths -----------------------------------------------------
    {
        const int total  = (int)(B * P);
        const int blocks = (total + THREADS - 1) / THREADS;
        dp_lengths_kernel<<<blocks, THREADS, 0, stream>>>(
            out_len, total, (int)P, (int)avg, (int)pb, (int)S);
    }

    // ---- patch_ids ---------------------------------------------------------
    const bool grouped = (S % 4 == 0) && (avg % 4 == 0);
    if (grouped) {
        const unsigned s_groups = (unsigned)(S / 4);     // b128 groups per row
        dim3 grid((s_groups + THREADS - 1) / THREADS, (unsigned)B, 1);
        dim3 block(THREADS, 1, 1);

        if (avg % BLOCK_ELEMS == 0) {
            // Block-uniform id: SALU-only value math (this instance: avg=1024).
            const unsigned bpp = (unsigned)(avg / BLOCK_ELEMS);
            dp_ids_uniform_kernel<<<grid, block, 0, stream>>>(
                (int4v*)out_ids, s_groups, bpp, (unsigned)pbc);
        } else if ((avg & (avg - 1)) == 0) {
            // Pow2 avg: shift instead of divide.
            unsigned sh = 0;
            while ((1LL << sh) < avg) ++sh;
            dp_ids_shift_kernel<<<grid, block, 0, stream>>>(
                (int4v*)out_ids, s_groups, sh, (unsigned)pbc);
        } else {
            dp_ids_b128_kernel<<<grid, block, 0, stream>>>(
                (int4v*)out_ids, s_groups, (unsigned)avg, (unsigned)pbc);
        }
    } else {
        const unsigned total  = (unsigned)(B * S);
        const unsigned blocks = (total + THREADS - 1) / THREADS;
        dp_ids_scalar_kernel<<<blocks, THREADS, 0, stream>>>(
            out_ids, total, (unsigned)S, (unsigned)avg, (unsigned)pbc);
    }
}